// Extrema1D_43800076484725
// MI455X (gfx1250) — compile-verified
//
#include <hip/hip_runtime.h>
#include <math.h>

// Problem constants from the reference: B=128, C=1, L=4096, d=32 (d read from input).
#define E_L 4096
#define E_T 256            // threads per block = 8 wave32
#define E_NC (E_L / 64)    // 64 chunks of 64 elements for the 2-level argmax

#if defined(__has_builtin)
#  if __has_builtin(__builtin_amdgcn_global_load_async_to_lds_b128)
#    define HAVE_ASYNC_LDS_LOAD 1
#  endif
#  if __has_builtin(__builtin_amdgcn_global_store_async_from_lds_b128)
#    define HAVE_ASYNC_LDS_STORE 1
#  endif
#  if __has_builtin(__builtin_amdgcn_s_wait_asynccnt)
#    define HAVE_WAIT_ASYNC 1
#  endif
#endif

// Async builtins take pointers to int4 (vector_size(16)) in global/LDS addr spaces.
typedef int v4i __attribute__((vector_size(16)));
typedef __attribute__((address_space(1))) v4i* g_v4i_ptr;  // global
typedef __attribute__((address_space(3))) v4i* l_v4i_ptr;  // LDS

// wave32 argmax butterfly; smallest-index tie-break (matches stable argsort).
// After this, ALL lanes hold the winner -> wave-uniform, no LDS broadcast needed.
__device__ __forceinline__ void waveArgmax(float &v, int &i) {
#pragma unroll
  for (int m = 16; m > 0; m >>= 1) {
    float ov = __shfl_xor(v, m, 32);
    int   oi = __shfl_xor(i, m, 32);
    if (ov > v || (ov == v && oi < i)) { v = ov; i = oi; }
  }
}

__global__ __launch_bounds__(E_T) void extrema1d_nms_kernel(
    const float* __restrict__ x, const int* __restrict__ dist,
    float* __restrict__ out) {
  __shared__ __attribute__((aligned(16))) float s_x[E_L];    // row data
  __shared__ __attribute__((aligned(16))) float s_key[E_L];  // candidate keys; reused as output staging
  __shared__ unsigned s_keep[E_L / 32];
  __shared__ float    s_cv[E_NC];   // per-chunk max key
  __shared__ int      s_ci[E_NC];   // per-chunk argmax (global index)

  const int    row  = blockIdx.x;
  const int    tid  = threadIdx.x;
  const float* xrow = x + (size_t)row * E_L;
  const int    d    = *dist;
  const float  NEG_INF = -__builtin_inff();

  __builtin_prefetch(xrow, 0, 3);  // global_prefetch_b8 on gfx1250

  // ---- Stage the row into LDS (CDNA5 async global->LDS DMA) ----
#ifdef HAVE_ASYNC_LDS_LOAD
  for (int k = tid; k < E_L / 4; k += E_T) {
    __builtin_amdgcn_global_load_async_to_lds_b128(
        (g_v4i_ptr)(xrow + 4 * k), (l_v4i_ptr)(&s_x[4 * k]), 0, 0);
  }
#  ifdef HAVE_WAIT_ASYNC
  __builtin_amdgcn_s_wait_asynccnt(0);
#  else
  asm volatile("s_wait_asynccnt 0" ::: "memory");
#  endif
#else
  for (int k = tid; k < E_L / 4; k += E_T) {
    ((float4*)s_x)[k] = ((const float4*)xrow)[k];
  }
#endif
  if (tid < E_L / 32) s_keep[tid] = 0u;
  __syncthreads();

  // ---- Extrema mask -> candidate keys ----
  // dxr[i] = (i<L-1) && (x[i+1]-x[i] > 0)   (right-pad 0 => false at L-1)
  // dxl[i] = (i==0) || (x[i]-x[i-1] <= 0)   (left-pad 0  => true at 0)
  // neg[i] = !(x[i] > 0)                    (1-sign(x) != 0; NaN => true)
  for (int i = tid; i < E_L; i += E_T) {
    float xi  = s_x[i];
    bool  dxr = (i < E_L - 1) && (s_x[i + 1] - xi > 0.0f);
    bool  dxl = (i == 0) || (xi - s_x[i - 1] <= 0.0f);
    bool  neg = !(xi > 0.0f);
    bool  valley = dxr && dxl && neg;
    bool  peak   = (!dxr) && (!dxl) && (!neg);
    s_key[i] = (valley || peak) ? fabsf(xi) : NEG_INF;
  }
  __syncthreads();

  // ---- Initial per-chunk maxima (wave per chunk; all 8 waves) ----
  {
    const int w = tid >> 5, lane = tid & 31;
    for (int c = w; c < E_NC; c += E_T / 32) {
      int   k = (c << 6) + lane;
      float v = s_key[k];      int idx = k;
      float vb = s_key[k + 32];
      if (vb > v) { v = vb; idx = k + 32; }   // k < k+32: keep lower index on ties
      waveArgmax(v, idx);
      if (lane == 0) { s_cv[c] = v; s_ci[c] = idx; }
    }
  }
  __syncthreads();

  // ---- Greedy NMS: single-wave loop, zero barriers ----
  // Equivalent to the reference's descending-|x| sweep: repeatedly take the
  // global argmax of live keys, keep it, kill keys in [p-d, p+d], and patch
  // only the <=2 chunk maxima the window touched. Kept points are pairwise
  // > d apart => <= L/(d+1)+1 iterations (125 for d=32). Intra-wave LDS ops
  // are in-order (DScnt), so no workgroup barriers are needed here.
  if (tid < 32) {
    const int lane = tid;
    for (int it = 0; it < E_L; ++it) {
      float bv = s_cv[lane];      int bi = s_ci[lane];
      float v2 = s_cv[lane + 32]; int i2 = s_ci[lane + 32];
      if (v2 > bv || (v2 == bv && i2 < bi)) { bv = v2; bi = i2; }
      waveArgmax(bv, bi);                 // all lanes now agree
      if (!(bv > NEG_INF)) break;         // no live extrema left
      const int p = bi;
      if (lane == 0) s_keep[p >> 5] |= (1u << (p & 31));
      int lo = p - d; if (lo < 0) lo = 0;
      int hi = p + d; if (hi > E_L - 1) hi = E_L - 1;
      for (int j = lo + lane; j <= hi; j += 32) s_key[j] = NEG_INF;
      const int clo = lo >> 6, chi = hi >> 6;
      for (int c = clo; c <= chi; ++c) {  // rescan touched chunks
        int   k = (c << 6) + lane;
        float v = s_key[k];      int idx = k;
        float vb = s_key[k + 32];
        if (vb > v) { v = vb; idx = k + 32; }
        waveArgmax(v, idx);
        if (lane == 0) { s_cv[c] = v; s_ci[c] = idx; }
      }
    }
  }
  __syncthreads();

  // ---- Masked output staged in LDS, then async LDS->global DMA ----
  for (int i = tid; i < E_L; i += E_T) {
    unsigned kb = (s_keep[i >> 5] >> (i & 31)) & 1u;
    s_key[i] = kb ? s_x[i] : 0.0f;
  }
  __syncthreads();
  float* orow = out + (size_t)row * E_L;
#ifdef HAVE_ASYNC_LDS_STORE
  for (int k = tid; k < E_L / 4; k += E_T) {
    __builtin_amdgcn_global_store_async_from_lds_b128(
        (g_v4i_ptr)(orow + 4 * k), (l_v4i_ptr)(&s_key[4 * k]), 0, 0);
  }
#  ifdef HAVE_WAIT_ASYNC
  __builtin_amdgcn_s_wait_asynccnt(0);
#  else
  asm volatile("s_wait_asynccnt 0" ::: "memory");
#  endif
#else
  for (int k = tid; k < E_L / 4; k += E_T) {
    ((float4*)orow)[k] = ((const float4*)s_key)[k];
  }
#endif
}

extern "C" void kernel_launch(void* const* d_in, const int* in_sizes, int n_in,
                              void* d_out, int out_size, void* d_ws, size_t ws_size,
                              hipStream_t stream) {
  const float* x    = (const float*)d_in[0];
  const int*   dptr = (const int*)d_in[1];   // minimum_extrema_distance (scalar)
  int B = in_sizes[0] / E_L;                 // 128 rows of length 4096
  extrema1d_nms_kernel<<<dim3(B), dim3(E_T), 0, stream>>>(x, dptr, (float*)d_out);
}